// SatelliteSpecificNormalization_23072564314709
// MI455X (gfx1250) — compile-verified
//
#include <hip/hip_runtime.h>
#include <hip/hip_bf16.h>

// ---------------------------------------------------------------------------
// SatelliteSpecificNormalization for MI455X (gfx1250).
//
// Pure streaming op: out = x * w[sid] + b[sid], 256 MiB total traffic,
// AI = 0.125 FLOP/B  ->  HBM-bound, floor ~11.5 us @ 23.3 TB/s.
//
// Strategy:
//   * one block per (image, chunk); satellite id / weight / bias are
//     wave-uniform scalars (scalar loads, no per-lane gather)
//   * CDNA5 async global->LDS b128 staging pipeline (ASYNCcnt-tracked),
//     4 stages deep per wave, no barriers (per-wave-private LDS slices)
//   * 128-bit accesses everywhere, nontemporal stores (stream-once data,
//     keep the 192MB L2 from being churned by the write stream)
//   * 2048 blocks x 8 wave32 = 16K waves, ~2KB async-in-flight per wave
//     -> ~32MB in flight device-wide (Little's law needs ~16MB @ 23.3TB/s)
// ---------------------------------------------------------------------------

typedef __attribute__((ext_vector_type(4))) float v4f;
typedef __attribute__((ext_vector_type(4))) int   v4i;

// Builtin signature (from hipcc diagnostic): first param is `int4 __device__*`
// i.e. v4i in address space 1 (global), non-const. LDS side is AS3.
typedef __attribute__((address_space(1))) v4i* as1_v4i_ptr;
typedef __attribute__((address_space(3))) v4i* as3_v4i_ptr;

#define THREADS   256
#define IMG_ELEMS (512 * 512)            // C == 1
#define VEC4_PER_IMG (IMG_ELEMS / 4)     // 65536 float4 per image
#define CHUNKS    16                     // blocks per image
#define ITERS     (VEC4_PER_IMG / (CHUNKS * THREADS))  // 16 float4-tiles/thread
#define STAGES    4                      // async pipeline depth (power of 2)
#define NUM_SAT   8

#if defined(__gfx1250__) && __has_builtin(__builtin_amdgcn_global_load_async_to_lds_b128)
#define HAVE_ASYNC_LDS 1
#else
#define HAVE_ASYNC_LDS 0
#endif

#if __has_builtin(__builtin_amdgcn_s_wait_asynccnt)
#define WAIT_ASYNC(n) __builtin_amdgcn_s_wait_asynccnt(n)
#else
#define WAIT_ASYNC(n) asm volatile("s_wait_asynccnt " #n ::: "memory")
#endif

#define WAIT_DS0() asm volatile("s_wait_dscnt 0" ::: "memory")

__global__ __launch_bounds__(THREADS) void
SatelliteSpecificNormalization_kernel(const float* __restrict__ x,
                                      const float* __restrict__ weight,
                                      const float* __restrict__ bias,
                                      const int*  __restrict__ sat,
                                      float* __restrict__ out)
{
    const int img   = blockIdx.y;     // flat (b,n) image index
    const int chunk = blockIdx.x;     // chunk within the image
    const int tid   = threadIdx.x;

    // Wave-uniform affine params (scalar path; sid uniform per block).
    const int  sid   = sat[img];
    const bool valid = (sid >= 0) && (sid < NUM_SAT);
    const int  sc    = valid ? sid : 0;
    const float w = valid ? weight[sc] : 1.0f;   // invalid -> identity
    const float b = valid ? bias[sc]   : 0.0f;

    const long base = (long)img * IMG_ELEMS + (long)chunk * (ITERS * THREADS * 4);
    const float* __restrict__ src = x   + base;
    float*       __restrict__ dst = out + base;

#define ELEM_OFF(it) ((long)(((it) * THREADS) + tid) * 4)

#if HAVE_ASYNC_LDS
    // Per-wave-private staging buffer: STAGES slots of THREADS*16B.
    __shared__ float lds[STAGES * THREADS * 4];   // 16 KB

    auto async_load = [&](int it, int s) {
        __builtin_amdgcn_global_load_async_to_lds_b128(
            (as1_v4i_ptr)(src + ELEM_OFF(it)),
            (as3_v4i_ptr)&lds[(s * THREADS + tid) * 4],
            /*imm offset*/ 0, /*cpol*/ 0);
    };
    auto consume = [&](int it) {
        const int s = it & (STAGES - 1);
        v4f v = *(const v4f*)&lds[(s * THREADS + tid) * 4];
        v4f r = v * w + b;
        __builtin_nontemporal_store(r, (v4f*)(dst + ELEM_OFF(it)));
    };

    // Prologue: fill the pipeline (ASYNCcnt-tracked, zero VGPR dest cost).
#pragma unroll
    for (int s = 0; s < STAGES; ++s)
        async_load(s, s);

    // Steady state: oldest load done <=> ASYNCcnt <= STAGES-1 (in-order).
    for (int it = 0; it < ITERS - STAGES; ++it) {
        const int s = it & (STAGES - 1);
        WAIT_ASYNC(3);                                   // STAGES-1
        v4f v = *(const v4f*)&lds[(s * THREADS + tid) * 4];
        WAIT_DS0();                // LDS read landed in VGPRs -> slot reusable
        async_load(it + STAGES, s);
        v4f r = v * w + b;
        __builtin_nontemporal_store(r, (v4f*)(dst + ELEM_OFF(it)));
    }

    // Drain: outstanding = 4,3,2,1 -> wait 3,2,1,0.
    WAIT_ASYNC(3); consume(ITERS - 4);
    WAIT_ASYNC(2); consume(ITERS - 3);
    WAIT_ASYNC(1); consume(ITERS - 2);
    WAIT_ASYNC(0); consume(ITERS - 1);
#else
    // Fallback: direct nontemporal 128-bit streaming (still HBM-optimal,
    // just holds in-flight loads in VGPRs instead of ASYNCcnt).
#pragma unroll 4
    for (int it = 0; it < ITERS; ++it) {
        v4f v = __builtin_nontemporal_load((const v4f*)(src + ELEM_OFF(it)));
        v4f r = v * w + b;
        __builtin_nontemporal_store(r, (v4f*)(dst + ELEM_OFF(it)));
    }
#endif
#undef ELEM_OFF
}

extern "C" void kernel_launch(void* const* d_in, const int* in_sizes, int n_in,
                              void* d_out, int out_size, void* d_ws, size_t ws_size,
                              hipStream_t stream) {
    // setup_inputs order: x (f32), weight (f32), bias (f32), satellite_ids (int)
    const float* x      = (const float*)d_in[0];
    const float* weight = (const float*)d_in[1];
    const float* bias   = (const float*)d_in[2];
    const int*   sat    = (const int*)d_in[3];
    float*       out    = (float*)d_out;

    const int numImages = in_sizes[3];          // 8*16 = 128
    dim3 grid(CHUNKS, (unsigned)numImages, 1);  // 16 x 128 = 2048 blocks
    SatelliteSpecificNormalization_kernel<<<grid, THREADS, 0, stream>>>(
        x, weight, bias, sat, out);
}